// NodeBlockDGL_42777874268720
// MI455X (gfx1250) — compile-verified
//
#include <hip/hip_runtime.h>
#include <hip/hip_bf16.h>

// Problem constants (match reference)
#define N_NODES 25000
#define N_EDGES 400000
#define DD      512        // node/edge feature dim
#define TWO_D   1024
#define HH      512        // hidden dim
#define LN_EPS  1e-5f
#define ROWS    32         // node rows per block (two 16-row WMMA M-tiles)

typedef __attribute__((ext_vector_type(2))) float v2f;
typedef __attribute__((ext_vector_type(8))) float v8f;

// ---------------- zero the aggregation buffer ----------------
__global__ void nb_zero_kernel(float* __restrict__ p, int n4) {
    int i = blockIdx.x * blockDim.x + threadIdx.x;
    if (i < n4) ((float4*)p)[i] = make_float4(0.f, 0.f, 0.f, 0.f);
}

// ---------------- scatter-add edges into agg -----------------
// One block per edge; 256 threads x 2 contiguous floats = 512 features.
__global__ void nb_scatter_kernel(const float* __restrict__ efeat,
                                  const int* __restrict__ edge_dst,
                                  float* __restrict__ agg) {
    const int e = blockIdx.x;
    const int t = threadIdx.x;
    const int d = edge_dst[e];
    const float2 v = ((const float2*)(efeat + (size_t)e * DD))[t];
    float* base = agg + (size_t)d * DD + t * 2;
    unsafeAtomicAdd(base,     v.x);   // global_atomic_add_f32
    unsafeAtomicAdd(base + 1, v.y);
}

__device__ __forceinline__ float fast_silu(float x) {
    // x * sigmoid(x); v_rcp_f32 instead of the IEEE-divide expansion.
    return x * __builtin_amdgcn_rcpf(1.0f + __expf(-x));
}

// ---------------- fused MLP + LayerNorm + residual -----------
// Block = 32 node rows, 256 threads = 8 waves (wave32).
// Each wave owns 4 col-tiles; dual accumulators (rows 0-15 / 16-31) so every
// B-fragment load feeds two v_wmma_f32_16x16x4_f32.
// Dynamic LDS: CAT 32x1024 (aliased as OUT 32x512 after GEMM1) | H 32x512 | RED 576
__launch_bounds__(256)
__global__ void nb_mlp_kernel(const float* __restrict__ nfeat,
                              const float* __restrict__ agg,
                              const float* __restrict__ w1,
                              const float* __restrict__ b1,
                              const float* __restrict__ w2,
                              const float* __restrict__ b2,
                              const float* __restrict__ ln_g,
                              const float* __restrict__ ln_b,
                              float* __restrict__ out) {
    extern __shared__ float smem[];
    float* CAT = smem;                     // 32*1024 floats (128KB)
    float* Hs  = smem + ROWS * TWO_D;      // 32*512 floats   (64KB)
    float* RED = Hs + ROWS * HH;           // 576 floats

    const int tid  = threadIdx.x;
    const int row0 = blockIdx.x * ROWS;

    // ---- load cat tile = [agg | nfeat] into LDS (float4, coalesced) ----
    #pragma unroll
    for (int j = 0; j < 32; ++j) {
        const int i4   = tid + j * 256;     // 0..8191 float4 slots
        const int r    = i4 >> 8;           // tile row (1024 floats = 256 f4/row)
        const int c4   = i4 & 255;
        const int grow = row0 + r;
        float4 v = make_float4(0.f, 0.f, 0.f, 0.f);
        if (grow < N_NODES) {
            if (c4 < 128) v = ((const float4*)(agg   + (size_t)grow * DD))[c4];
            else          v = ((const float4*)(nfeat + (size_t)grow * DD))[c4 - 128];
        }
        ((float4*)CAT)[i4] = v;
    }
    __syncthreads();

    const int lane = tid & 31;
    const int wave = tid >> 5;
    const int nl   = lane & 15;            // N column within 16-wide tile
    const int koff = (lane >> 4) << 1;     // 0 for lanes 0-15, 2 for lanes 16-31
    const int mhi  = (lane >> 4) << 3;     // C/D row offset: 0 or 8

    // ---- GEMM1: cat[32x1024] @ w1[1024x512] + b1, SiLU -> Hs ----
    for (int t = 0; t < 4; ++t) {
        const int n0 = (wave * 4 + t) * 16;
        v8f acc0, acc1;
        const float bias1 = b1[n0 + nl];
        #pragma unroll
        for (int j = 0; j < 8; ++j) { acc0[j] = bias1; acc1[j] = bias1; }

        const float* ap0  = CAT + nl * TWO_D + koff;          // rows 0..15
        const float* ap1  = CAT + (nl + 16) * TWO_D + koff;   // rows 16..31
        const float* bcol = w1 + n0 + nl;                     // row-major, lane = col
        #pragma unroll 4
        for (int k = 0; k < TWO_D; k += 4) {
            v2f b;  b.x  = bcol[(size_t)(k + koff) * HH];
                    b.y  = bcol[(size_t)(k + koff + 1) * HH];
            v2f a0; a0.x = ap0[k]; a0.y = ap0[k + 1];
            v2f a1; a1.x = ap1[k]; a1.y = ap1[k + 1];
            acc0 = __builtin_amdgcn_wmma_f32_16x16x4_f32(
                       false, a0, false, b, (short)0, acc0, false, false);
            acc1 = __builtin_amdgcn_wmma_f32_16x16x4_f32(
                       false, a1, false, b, (short)0, acc1, false, false);
        }
        #pragma unroll
        for (int j = 0; j < 8; ++j) {
            const int m = j + mhi;
            Hs[m * HH + n0 + nl]        = fast_silu(acc0[j]);
            Hs[(m + 16) * HH + n0 + nl] = fast_silu(acc1[j]);
        }
    }
    __syncthreads();

    // ---- GEMM2: Hs[32x512] @ w2[512x512] + b2 -> OUT (aliases CAT) ----
    float* OUT = CAT;
    for (int t = 0; t < 4; ++t) {
        const int n0 = (wave * 4 + t) * 16;
        v8f acc0, acc1;
        const float bias2 = b2[n0 + nl];
        #pragma unroll
        for (int j = 0; j < 8; ++j) { acc0[j] = bias2; acc1[j] = bias2; }

        const float* ap0  = Hs + nl * HH + koff;
        const float* ap1  = Hs + (nl + 16) * HH + koff;
        const float* bcol = w2 + n0 + nl;
        #pragma unroll 4
        for (int k = 0; k < HH; k += 4) {
            v2f b;  b.x  = bcol[(size_t)(k + koff) * DD];
                    b.y  = bcol[(size_t)(k + koff + 1) * DD];
            v2f a0; a0.x = ap0[k]; a0.y = ap0[k + 1];
            v2f a1; a1.x = ap1[k]; a1.y = ap1[k + 1];
            acc0 = __builtin_amdgcn_wmma_f32_16x16x4_f32(
                       false, a0, false, b, (short)0, acc0, false, false);
            acc1 = __builtin_amdgcn_wmma_f32_16x16x4_f32(
                       false, a1, false, b, (short)0, acc1, false, false);
        }
        #pragma unroll
        for (int j = 0; j < 8; ++j) {
            const int m = j + mhi;
            OUT[m * DD + n0 + nl]        = acc0[j];
            OUT[(m + 16) * DD + n0 + nl] = acc1[j];
        }
    }
    __syncthreads();

    // ---- LayerNorm(row of 512) * gamma + beta + residual ----
    // 256 threads = 32 rows x 8 groups of 64 contiguous cols.
    const int r = tid >> 3;
    const int g = tid & 7;
    const float* orow = OUT + r * DD + g * 64;
    float s = 0.f, ss = 0.f;
    #pragma unroll 8
    for (int c = 0; c < 64; ++c) { const float x = orow[c]; s += x; ss += x * x; }
    RED[r * 8 + g]       = s;
    RED[256 + r * 8 + g] = ss;
    __syncthreads();
    if (g == 0) {
        float S = 0.f, SS = 0.f;
        #pragma unroll
        for (int i = 0; i < 8; ++i) { S += RED[r * 8 + i]; SS += RED[256 + r * 8 + i]; }
        const float mu  = S * (1.0f / DD);
        const float var = SS * (1.0f / DD) - mu * mu;
        RED[512 + r] = mu;
        RED[544 + r] = rsqrtf(var + LN_EPS);
    }
    __syncthreads();
    const int grow = row0 + r;
    if (grow < N_NODES) {
        const float mu = RED[512 + r];
        const float rs = RED[544 + r];
        const float* nf = nfeat + (size_t)grow * DD + g * 64;
        float*       op = out   + (size_t)grow * DD + g * 64;
        const float* gm = ln_g + g * 64;
        const float* bt = ln_b + g * 64;
        #pragma unroll 8
        for (int c = 0; c < 64; ++c)
            op[c] = (orow[c] - mu) * rs * gm[c] + bt[c] + nf[c];
    }
}

extern "C" void kernel_launch(void* const* d_in, const int* in_sizes, int n_in,
                              void* d_out, int out_size, void* d_ws, size_t ws_size,
                              hipStream_t stream) {
    const float* efeat = (const float*)d_in[0];
    const float* nfeat = (const float*)d_in[1];
    const int*   edst  = (const int*)  d_in[2];
    const float* w1    = (const float*)d_in[3];
    const float* b1    = (const float*)d_in[4];
    const float* w2    = (const float*)d_in[5];
    const float* b2    = (const float*)d_in[6];
    const float* ln_g  = (const float*)d_in[7];
    const float* ln_b  = (const float*)d_in[8];

    float* out       = (float*)d_out;
    float* out_nfeat = out + (size_t)N_EDGES * DD;   // second tuple element
    // out_nfeat doubles as the segment-sum accumulator (exactly N_NODES*D f32);
    // each MLP block reads its own rows before overwriting them -> safe aliasing.

    // 1) efeat pass-through (first tuple element), 819 MB d2d
    hipMemcpyAsync(out, efeat, (size_t)N_EDGES * DD * sizeof(float),
                   hipMemcpyDeviceToDevice, stream);

    // 2) zero agg
    nb_zero_kernel<<<(N_NODES * DD / 4 + 255) / 256, 256, 0, stream>>>(
        out_nfeat, N_NODES * DD / 4);

    // 3) scatter-add edge features
    nb_scatter_kernel<<<N_EDGES, 256, 0, stream>>>(efeat, edst, out_nfeat);

    // 4) fused MLP + LayerNorm + residual (32 rows / block)
    const size_t lds_bytes =
        (size_t)(ROWS * TWO_D + ROWS * HH + 576) * sizeof(float);
    nb_mlp_kernel<<<(N_NODES + ROWS - 1) / ROWS, 256, lds_bytes, stream>>>(
        nfeat, out_nfeat, w1, b1, w2, b2, ln_g, ln_b, out_nfeat);
}